// CNNTopKMoE_28518582845695
// MI455X (gfx1250) — compile-verified
//
#include <hip/hip_runtime.h>

#define BN_EPS 1e-5f

typedef __attribute__((ext_vector_type(2))) float v2f;
typedef __attribute__((ext_vector_type(8))) float v8f;

// D = A(16x4,f32) * B(4x16,f32) + C(16x16,f32)
__device__ __forceinline__ v8f wmma4(v2f a, v2f b, v8f c) {
  return __builtin_amdgcn_wmma_f32_16x16x4_f32(false, a, false, b, (short)0, c,
                                               false, false);
}

// ---------------------------------------------------------------------------
// Async global->LDS staging (CDNA5 GLOBAL_LOAD_ASYNC_TO_LDS_B128, ASYNCcnt).
// Builtin expects v4i pointers: global source in AS(1), LDS dest in AS(3).
// Casts go through integers: generic LDS pointer low 32 bits == LDS offset
// (ISA aperture rule), global generic == AS1 numerically.
// ---------------------------------------------------------------------------
#if defined(__AMDGCN__) && __has_builtin(__builtin_amdgcn_global_load_async_to_lds_b128)
#define HAVE_ASYNC_LDS 1
typedef __attribute__((ext_vector_type(4))) int v4i;
typedef __attribute__((address_space(1))) v4i as1_v4i;
typedef __attribute__((address_space(3))) v4i as3_v4i;
__device__ __forceinline__ void async_cp16(const float* g, float* l) {
  __builtin_amdgcn_global_load_async_to_lds_b128(
      (as1_v4i*)(unsigned long long)g,
      (as3_v4i*)(unsigned int)(unsigned long long)l, 0, 0);
}
#define ASYNC_WAIT() asm volatile("s_wait_asynccnt 0x0" ::: "memory")
#else
#define HAVE_ASYNC_LDS 0
__device__ __forceinline__ void async_cp16(const float* g, float* l) {
  *(float4*)l = *(const float4*)g;   // fallback: copy through VGPRs
}
#define ASYNC_WAIT()
#endif

// Copy N floats (N multiple of 1024) from contiguous global to LDS, 256 thr.
template <int N>
__device__ __forceinline__ void stage_chunk(const float* g, float* l, int tid) {
  #pragma unroll
  for (int i = 0; i < N / 1024; ++i)
    async_cp16(g + 4 * tid + i * 1024, l + 4 * tid + i * 1024);
}

// ---------------------------------------------------------------------------
// Prep: transpose conv weights to [tap][cin][cout] so each tap chunk is one
// contiguous block for wide async staging.
// ---------------------------------------------------------------------------
__global__ __launch_bounds__(256) void prep_kernel(
    const float* __restrict__ c2w, const float* __restrict__ c3w,
    float* __restrict__ c2t, float* __restrict__ c3t)
{
  const int i = blockIdx.x * 256 + threadIdx.x;
  if (i < 18432) {                       // c2t[pos][c(32)][n(64)]
    int pos = i >> 11, r = i & 2047, c = r >> 6, n = r & 63;
    c2t[i] = c2w[(n * 32 + c) * 9 + pos];
  }
  if (i < 73728) {                       // c3t[pos][c(64)][n(128)]
    int pos = i >> 13, r = i & 8191, c = r >> 7, n = r & 127;
    c3t[i] = c3w[(n * 64 + c) * 9 + pos];
  }
}

// ---------------------------------------------------------------------------
// Fused backbone: conv1(VALU) -> conv2(WMMA) -> conv3(WMMA) -> feat[b,128]
// One block (256 thr = 8 waves) per image; all activations stay in LDS.
// LDS regions (floats):
//   h1  [18][18][32] halo                         10368 @ 0
//   rB  (x [3][34][34] ph1 / h2 [10][10][64])      6400 @ 10368
//   rC  double-buffered weight staging            16384 @ 16768
// ---------------------------------------------------------------------------
#define OFF_B  10368
#define OFF_C  16768
#define SMEM_F 33152

__global__ __launch_bounds__(256) void backbone_kernel(
    const float* __restrict__ x,
    const float* __restrict__ c1w, const float* __restrict__ c1b,
    const float* __restrict__ bn1g, const float* __restrict__ bn1b,
    const float* __restrict__ bn1m, const float* __restrict__ bn1v,
    const float* __restrict__ c2t, const float* __restrict__ c2b,
    const float* __restrict__ bn2g, const float* __restrict__ bn2b,
    const float* __restrict__ bn2m, const float* __restrict__ bn2v,
    const float* __restrict__ c3t, const float* __restrict__ c3b,
    const float* __restrict__ bn3g, const float* __restrict__ bn3b,
    const float* __restrict__ bn3m, const float* __restrict__ bn3v,
    float* __restrict__ feat)
{
  __shared__ __align__(16) float smem[SMEM_F];
  const int tid  = threadIdx.x;
  const int b    = blockIdx.x;
  const int lane = tid & 31;
  const int wid  = tid >> 5;
  const int colL = lane & 15;   // A row (M) / B,C column (N)
  const int hl   = lane >> 4;
  const int koff = hl * 2;      // K offset within 4-wide K step

  float* h1 = &smem[0];
  float* rB = &smem[OFF_B];
  float* rC = &smem[OFF_C];

  // ---------------- Phase 1: conv1 3->32 + BN + ReLU + 2x2 pool -> h1 -----
  for (int i = tid; i < 10368; i += 256) h1[i] = 0.0f;
  for (int i = tid; i < 3468;  i += 256) rB[i] = 0.0f;
  {
    const float* xg = x + (size_t)b * 3072;
    for (int i = tid; i < 3072; i += 256) {
      int c = i >> 10, r = i & 1023, yy = r >> 5, xx = r & 31;
      rB[c * 1156 + (yy + 1) * 34 + (xx + 1)] = xg[i];
    }
  }
  for (int i = tid; i < 864; i += 256) rC[i] = c1w[i];
  if (tid < 32) {
    float inv = bn1g[tid] * rsqrtf(bn1v[tid] + BN_EPS);
    rC[1024 + tid] = inv;
    rC[1056 + tid] = bn1b[tid] + (c1b[tid] - bn1m[tid]) * inv;
  }
  __syncthreads();
  {
    const int yo = tid >> 4, xo = tid & 15;   // pooled output coords 16x16
    float patch[3][4][4];                      // 4x4 input patch, 3 channels
    #pragma unroll
    for (int ci = 0; ci < 3; ++ci)
      #pragma unroll
      for (int py = 0; py < 4; ++py)
        #pragma unroll
        for (int px = 0; px < 4; ++px)
          patch[ci][py][px] = rB[ci * 1156 + (2 * yo + py) * 34 + (2 * xo + px)];
    for (int oc = 0; oc < 32; ++oc) {
      float a0 = 0.f, a1 = 0.f, a2 = 0.f, a3 = 0.f;
      #pragma unroll
      for (int ci = 0; ci < 3; ++ci)
        #pragma unroll
        for (int ky = 0; ky < 3; ++ky)
          #pragma unroll
          for (int kx = 0; kx < 3; ++kx) {
            const float wv = rC[oc * 27 + ci * 9 + ky * 3 + kx];
            a0 += wv * patch[ci][ky][kx];
            a1 += wv * patch[ci][ky][kx + 1];
            a2 += wv * patch[ci][ky + 1][kx];
            a3 += wv * patch[ci][ky + 1][kx + 1];
          }
      const float aa = rC[1024 + oc], bb = rC[1056 + oc];
      float mx = fmaxf(fmaxf(fmaxf(aa * a0 + bb, 0.f), fmaxf(aa * a1 + bb, 0.f)),
                       fmaxf(fmaxf(aa * a2 + bb, 0.f), fmaxf(aa * a3 + bb, 0.f)));
      h1[((yo + 1) * 18 + (xo + 1)) * 32 + oc] = mx;
    }
  }
  __syncthreads();

  // zero h2 region + pre-issue conv2 tap chunk 0 (async, overlaps zeroing)
  for (int i = tid; i < 6400; i += 256) rB[i] = 0.0f;
  stage_chunk<2048>(c2t, rC, tid);

  // ---------- Phase 2: conv2 implicit GEMM  M=256 px, N=64, K=288 --------
  {
    const int nt    = wid & 3;
    const int ybase = (wid >> 2) * 8;
    v8f acc[8];
    #pragma unroll
    for (int t = 0; t < 8; ++t) acc[t] = {};

    for (int pos = 0; pos < 9; ++pos) {
      ASYNC_WAIT();
      __syncthreads();
      if (pos < 8)  // prefetch next tap chunk into the other buffer
        stage_chunk<2048>(c2t + (pos + 1) * 2048, rC + ((pos + 1) & 1) * 2048, tid);
      const float* wch = rC + (pos & 1) * 2048;
      const int ky = pos / 3, kx = pos - 3 * ky;
      const int xi = colL + kx;
      #pragma unroll
      for (int s = 0; s < 8; ++s) {
        const int c0 = 4 * s + koff;
        v2f bf;
        bf.x = wch[c0 * 64 + nt * 16 + colL];
        bf.y = wch[(c0 + 1) * 64 + nt * 16 + colL];
        #pragma unroll
        for (int t = 0; t < 8; ++t) {
          const int idx = ((ybase + t + ky) * 18 + xi) * 32 + c0;
          v2f af; af.x = h1[idx]; af.y = h1[idx + 1];
          acc[t] = wmma4(af, bf, acc[t]);
        }
      }
    }
    // epilogue: BN + ReLU + 2x2 maxpool -> h2[10][10][64]
    {
      const int ch = nt * 16 + colL;
      const float inv = bn2g[ch] * rsqrtf(bn2v[ch] + BN_EPS);
      const float sh  = bn2b[ch] + (c2b[ch] - bn2m[ch]) * inv;
      #pragma unroll
      for (int j = 0; j < 4; ++j) {
        #pragma unroll
        for (int i = 0; i < 4; ++i) {
          float v0 = fmaxf(inv * acc[2 * j][2 * i]     + sh, 0.0f);
          float v1 = fmaxf(inv * acc[2 * j][2 * i + 1] + sh, 0.0f);
          float v2 = fmaxf(inv * acc[2 * j + 1][2 * i]     + sh, 0.0f);
          float v3 = fmaxf(inv * acc[2 * j + 1][2 * i + 1] + sh, 0.0f);
          float p  = fmaxf(fmaxf(v0, v1), fmaxf(v2, v3));
          rB[(((ybase >> 1) + j + 1) * 10 + (hl * 4 + i + 1)) * 64 + ch] = p;
        }
      }
    }
  }
  __syncthreads();
  stage_chunk<8192>(c3t, rC, tid);   // conv3 tap chunk 0

  // ---------- Phase 3: conv3 implicit GEMM  M=64 px, N=128, K=576 --------
  {
    const int nt = wid;
    v8f acc[4];
    #pragma unroll
    for (int m = 0; m < 4; ++m) acc[m] = {};

    for (int pos = 0; pos < 9; ++pos) {
      ASYNC_WAIT();
      __syncthreads();
      if (pos < 8)
        stage_chunk<8192>(c3t + (pos + 1) * 8192, rC + ((pos + 1) & 1) * 8192, tid);
      const float* wch = rC + (pos & 1) * 8192;
      const int ky = pos / 3, kx = pos - 3 * ky;
      const int yA = (colL >> 3);       // pixel p = 16m + M; y = 2m + (M>>3)
      const int xA = (colL & 7) + kx;
      #pragma unroll
      for (int s = 0; s < 16; ++s) {
        const int c0 = 4 * s + koff;
        v2f bf;
        bf.x = wch[c0 * 128 + nt * 16 + colL];
        bf.y = wch[(c0 + 1) * 128 + nt * 16 + colL];
        #pragma unroll
        for (int m = 0; m < 4; ++m) {
          const int idx = ((2 * m + yA + ky) * 10 + xA) * 64 + c0;
          v2f af; af.x = rB[idx]; af.y = rB[idx + 1];
          acc[m] = wmma4(af, bf, acc[m]);
        }
      }
    }
    // epilogue: BN + ReLU + 2x2 pool (y-pairs across lane halves) + mean
    const int ch = nt * 16 + colL;
    const float inv = bn3g[ch] * rsqrtf(bn3v[ch] + BN_EPS);
    const float sh  = bn3b[ch] + (c3b[ch] - bn3m[ch]) * inv;
    float fs = 0.0f;
    #pragma unroll
    for (int m = 0; m < 4; ++m) {
      float mm[8];
      #pragma unroll
      for (int r = 0; r < 8; ++r) {
        float e = fmaxf(inv * acc[m][r] + sh, 0.0f);  // lane half: y = 2m+hl
        float o = __shfl_xor(e, 16, 32);
        mm[r] = fmaxf(e, o);
      }
      #pragma unroll
      for (int i = 0; i < 4; ++i)
        fs += fmaxf(mm[2 * i], mm[2 * i + 1]);
    }
    if (hl == 0) feat[(size_t)b * 128 + ch] = fs * (1.0f / 16.0f);
  }
}

// ---------------------------------------------------------------------------
// Gate: logits = feat @ gate_w + gate_b ; top-2 ; softmax ; scatter [B,8]
// ---------------------------------------------------------------------------
__global__ __launch_bounds__(256) void gate_kernel(
    const float* __restrict__ feat, const float* __restrict__ gw,
    const float* __restrict__ gb, float* __restrict__ wout)
{
  const int wid = threadIdx.x >> 5, lane = threadIdx.x & 31;
  const int row = blockIdx.x * 8 + wid;
  float l[8];
  #pragma unroll
  for (int e = 0; e < 8; ++e) l[e] = 0.0f;
  const float* fr = feat + (size_t)row * 128;
  for (int f = lane; f < 128; f += 32) {
    const float fv = fr[f];
    #pragma unroll
    for (int e = 0; e < 8; ++e) l[e] += fv * gw[f * 8 + e];
  }
  #pragma unroll
  for (int off = 16; off >= 1; off >>= 1)
    #pragma unroll
    for (int e = 0; e < 8; ++e) l[e] += __shfl_xor(l[e], off, 32);
  #pragma unroll
  for (int e = 0; e < 8; ++e) l[e] += gb[e];
  if (lane == 0) {
    int i1 = 0; float v1 = l[0];
    #pragma unroll
    for (int e = 1; e < 8; ++e) if (l[e] > v1) { v1 = l[e]; i1 = e; }
    int i2 = -1; float v2 = -1e30f;
    #pragma unroll
    for (int e = 0; e < 8; ++e) if (e != i1 && l[e] > v2) { v2 = l[e]; i2 = e; }
    const float e2 = __expf(v2 - v1);
    const float d  = 1.0f + e2;
    const float wa = 1.0f / d, wb = e2 / d;
    #pragma unroll
    for (int e = 0; e < 8; ++e)
      wout[(size_t)row * 8 + e] = (e == i1) ? wa : ((e == i2) ? wb : 0.0f);
  }
}

// ---------------------------------------------------------------------------
// Experts: out[b,:] = sum_e gate[b,e]*(relu(feat[b]@w1[e]+b1[e])@w2[e]+b2[e])
// Block = 128 rows (8 waves x 16-row M-tiles); both GEMMs on WMMA.
// w1[e] double-buffered via async LDS copies (overlaps previous expert).
// ---------------------------------------------------------------------------
#define E_FEAT 0          // 16384
#define E_W1   16384      // 2 x 8192
#define E_HE   32768      // 8 x 1024
#define E_W2   40960      // 1024
#define E_B1   41984      // 64
#define E_B2   42048      // 16
#define E_GW   42064      // 1024
#define E_SMEM 43088

__global__ __launch_bounds__(256) void expert_kernel(
    const float* __restrict__ feat, const float* __restrict__ gwts,
    const float* __restrict__ w1, const float* __restrict__ b1,
    const float* __restrict__ w2, const float* __restrict__ b2,
    float* __restrict__ out)
{
  __shared__ __align__(16) float smem[E_SMEM];
  const int tid  = threadIdx.x;
  const int lane = tid & 31, wid = tid >> 5;
  const int colL = lane & 15, hl = lane >> 4, koff = hl * 2;
  const int rbase = blockIdx.x * 128;

  float* fL  = &smem[E_FEAT];
  float* heL = &smem[E_HE + wid * 1024];   // per-wave 16x64 scratch
  float* w2L = &smem[E_W2];
  float* b1L = &smem[E_B1];
  float* b2L = &smem[E_B2];
  float* gwL = &smem[E_GW];

  stage_chunk<16384>(feat + (size_t)rbase * 128, fL, tid);
  stage_chunk<1024>(gwts + (size_t)rbase * 8, gwL, tid);
  stage_chunk<8192>(w1, &smem[E_W1], tid);          // expert 0 -> buffer 0

  v8f acc = {};
  for (int e = 0; e < 8; ++e) {
    ASYNC_WAIT();                 // w1[e] (and feat/gate on first pass) landed
    __syncthreads();
    for (int i = tid; i < 1024; i += 256) {
      int k = i >> 4, n = i & 15;
      w2L[i] = (n < 10) ? w2[e * 640 + k * 10 + n] : 0.0f;
    }
    if (tid < 64) b1L[tid] = b1[e * 64 + tid];
    if (tid < 16) b2L[tid] = (tid < 10) ? b2[e * 10 + tid] : 0.0f;
    if (e < 7)                    // prefetch next expert, overlaps compute
      stage_chunk<8192>(w1 + (e + 1) * 8192, &smem[E_W1 + ((e + 1) & 1) * 8192], tid);
    __syncthreads();
    const float* w1b = &smem[E_W1 + (e & 1) * 8192];

    // he = relu(feat @ w1[e] + b1[e])   (16 rows x 64, K=128)
    #pragma unroll
    for (int nt = 0; nt < 4; ++nt) {
      v8f h = {};
      #pragma unroll
      for (int s = 0; s < 32; ++s) {
        const int c0 = 4 * s + koff;
        v2f af; af.x = fL[(wid * 16 + colL) * 128 + c0];
                af.y = fL[(wid * 16 + colL) * 128 + c0 + 1];
        v2f bf; bf.x = w1b[c0 * 64 + nt * 16 + colL];
                bf.y = w1b[(c0 + 1) * 64 + nt * 16 + colL];
        h = wmma4(af, bf, h);
      }
      const float bias = b1L[nt * 16 + colL];
      #pragma unroll
      for (int r = 0; r < 8; ++r)
        heL[(r + 8 * hl) * 64 + nt * 16 + colL] = fmaxf(h[r] + bias, 0.0f);
    }
    // DS ops from one wave complete in order; fence the compiler only.
    __builtin_amdgcn_wave_barrier();

    // oe = he @ w2[e]   (16 rows x 16(pad), K=64)
    v8f o = {};
    #pragma unroll
    for (int s = 0; s < 16; ++s) {
      const int c0 = 4 * s + koff;
      v2f af; af.x = heL[colL * 64 + c0]; af.y = heL[colL * 64 + c0 + 1];
      v2f bf; bf.x = w2L[c0 * 16 + colL]; bf.y = w2L[(c0 + 1) * 16 + colL];
      o = wmma4(af, bf, o);
    }
    const float bo = b2L[colL];
    #pragma unroll
    for (int r = 0; r < 8; ++r)
      acc[r] += gwL[(wid * 16 + r + 8 * hl) * 8 + e] * (o[r] + bo);
  }
  if (colL < 10) {
    #pragma unroll
    for (int r = 0; r < 8; ++r) {
      const int grow = rbase + wid * 16 + r + 8 * hl;
      out[(size_t)grow * 10 + colL] = acc[r];
    }
  }
}

// ---------------------------------------------------------------------------
extern "C" void kernel_launch(void* const* d_in, const int* in_sizes, int n_in,
                              void* d_out, int out_size, void* d_ws,
                              size_t ws_size, hipStream_t stream)
{
  const float* x    = (const float*)d_in[0];
  const float* c1w  = (const float*)d_in[1];
  const float* c1b  = (const float*)d_in[2];
  const float* bn1g = (const float*)d_in[3];
  const float* bn1b = (const float*)d_in[4];
  const float* bn1m = (const float*)d_in[5];
  const float* bn1v = (const float*)d_in[6];
  const float* c2w  = (const float*)d_in[7];
  const float* c2b  = (const float*)d_in[8];
  const float* bn2g = (const float*)d_in[9];
  const float* bn2b = (const float*)d_in[10];
  const float* bn2m = (const float*)d_in[11];
  const float* bn2v = (const float*)d_in[12];
  const float* c3w  = (const float*)d_in[13];
  const float* c3b  = (const float*)d_in[14];
  const float* bn3g = (const float*)d_in[15];
  const float* bn3b = (const float*)d_in[16];
  const float* bn3m = (const float*)d_in[17];
  const float* bn3v = (const float*)d_in[18];
  const float* w1   = (const float*)d_in[19];
  const float* b1   = (const float*)d_in[20];
  const float* w2   = (const float*)d_in[21];
  const float* b2   = (const float*)d_in[22];
  const float* gw   = (const float*)d_in[23];
  const float* gb   = (const float*)d_in[24];

  float* feat = (float*)d_ws;                      // 4096*128
  float* gwts = feat + (size_t)4096 * 128;         // 4096*8
  float* c2t  = gwts + (size_t)4096 * 8;           // 9*32*64   = 18432
  float* c3t  = c2t + 18432;                       // 9*64*128  = 73728

  prep_kernel<<<288, 256, 0, stream>>>(c2w, c3w, c2t, c3t);

  backbone_kernel<<<4096, 256, 0, stream>>>(
      x, c1w, c1b, bn1g, bn1b, bn1m, bn1v,
      c2t, c2b, bn2g, bn2b, bn2m, bn2v,
      c3t, c3b, bn3g, bn3b, bn3m, bn3v, feat);

  gate_kernel<<<512, 256, 0, stream>>>(feat, gw, gb, gwts);

  expert_kernel<<<32, 256, 0, stream>>>(feat, gwts, w1, b1, w2, b2,
                                        (float*)d_out);
}